// decode_atten_44384192037516
// MI455X (gfx1250) — compile-verified
//
#include <hip/hip_runtime.h>
#include <hip/hip_bf16.h>

typedef __attribute__((ext_vector_type(2))) float v2f;
typedef __attribute__((ext_vector_type(8))) float v8f;

#define BATCH 8
#define CCH   16
#define HH    512
#define WW    512
// pooled: 64x64 per channel
// U2: K-pair-interleaved B matrix: U2[b][l'/2][n][2] (= 2MB total)
// attn: [B][4096][64]; out = x shape

// ---------------------------------------------------------------------------
// Kernel 1: 8x8 average pool + pack into K-pair-interleaved GEMM-B layout
//   l' = (ph>>3)*8 + (pw>>3),  n = c*64 + (ph&7)*8 + (pw&7)
//   U2 flat index = (l'>>1)*2048 + n*2 + (l'&1)
// ---------------------------------------------------------------------------
__global__ __launch_bounds__(256) void pool_pack_kernel(
    const float* __restrict__ x, float* __restrict__ U2)
{
    int idx = blockIdx.x * blockDim.x + threadIdx.x;  // B*C*64*64 = 524288
    int pw = idx & 63;
    int ph = (idx >> 6) & 63;
    int c  = (idx >> 12) & 15;
    int b  = idx >> 16;

    const float* src = x + (((size_t)(b * CCH + c) * HH) + (size_t)ph * 8) * WW + pw * 8;
    float s = 0.0f;
#pragma unroll
    for (int r = 0; r < 8; ++r) {
        const float4* p = (const float4*)(src + (size_t)r * WW);
        float4 a = p[0];
        float4 bb = p[1];
        s += (a.x + a.y) + (a.z + a.w) + (bb.x + bb.y) + (bb.z + bb.w);
    }
    s *= (1.0f / 64.0f);

    int lp = ((ph >> 3) << 3) | (pw >> 3);            // 0..63  (GEMM K index)
    int n  = (c << 6) | ((ph & 7) << 3) | (pw & 7);   // 0..1023 (GEMM N index)
    U2[(size_t)b * 65536 + (size_t)(lp >> 1) * 2048 + n * 2 + (lp & 1)] = s;
}

// ---------------------------------------------------------------------------
// Kernel 2: per-batch GEMM  O[4096x1024] = Ahat[4096x64] * U[64x1024]
// fused row nonzero-count normalization + folded-layout stores.
// One workgroup = 8 waves = one 16-row M-tile, full N=1024 (wave: 8 tiles).
// B fragments are single b64 loads thanks to K-pair interleave.
// ---------------------------------------------------------------------------
#define LDSS 68  // padded row stride (floats): conflict-free, keeps 16B align

__global__ __launch_bounds__(256) void att_gemm_kernel(
    const float* __restrict__ attn,   // [B][4096][64]
    const float* __restrict__ U2,     // [B][32][1024][2]
    float* __restrict__ out)          // [B][16][512][512]
{
    __shared__ float As[16 * LDSS];
    __shared__ float rowScale[16];

    int bid   = blockIdx.x;        // b*256 + mtile
    int mtile = bid & 255;
    int b     = bid >> 8;
    int l_base = mtile * 16;

    int t    = threadIdx.x;
    int lane = t & 31;
    int wave = t >> 5;

    // cooperative A-tile load: 16 rows x 64 cols, one float4 per thread (raw)
    {
        int row  = t >> 4;          // 0..15
        int col4 = (t & 15) * 4;    // 0..60
        float4 v = *(const float4*)(attn + ((size_t)b * 4096 + l_base + row) * 64 + col4);
        *(float4*)(&As[row * LDSS + col4]) = v;
    }
    __syncthreads();

    // fused nonzero-count normalization: scale = 1/(count + 1e-5)
    if (t < 16) {
        int cnt = 0;
#pragma unroll
        for (int k = 0; k < 64; ++k) cnt += (As[t * LDSS + k] != 0.0f) ? 1 : 0;
        rowScale[t] = 1.0f / ((float)cnt + 1e-5f);
    }
    __syncthreads();

    int n_wave = wave * 128;       // this wave's N base (8 tiles of 16)
    int M = lane & 15;             // A row / B col / D col index for this lane
    int h = lane >> 4;             // half-wave selector
    float scale = rowScale[M];

    v8f acc[8];
#pragma unroll
    for (int i = 0; i < 8; ++i) acc[i] = (v8f){0.f,0.f,0.f,0.f,0.f,0.f,0.f,0.f};

    const float* Ub = U2 + (size_t)b * 65536;

#pragma unroll
    for (int ks = 0; ks < 16; ++ks) {
        int k0 = ks * 4 + h * 2;   // this lane's K pair (always even)
        v2f a;
        a.x = As[M * LDSS + k0]     * scale;
        a.y = As[M * LDSS + k0 + 1] * scale;
#pragma unroll
        for (int nt = 0; nt < 8; ++nt) {
            int n = n_wave + nt * 16 + M;
            // one 8-byte load: (U[k0][n], U[k0+1][n])
            v2f bf = *(const v2f*)(Ub + (size_t)(k0 >> 1) * 2048 + n * 2);
            acc[nt] = __builtin_amdgcn_wmma_f32_16x16x4_f32(
                false, a, false, bf, (short)0, acc[nt], false, false);
        }
    }

    // Store with fold: l = l_base + (v + 8h); hp = l>>6 (constant in tile),
    // wp = l&63; k = n_wave + nt*16 + M -> c,kh,kw; out[b,c,hp*8+kh,wp*8+kw]
    int hp  = l_base >> 6;
    int wp0 = (l_base & 63) + 8 * h;   // wp for v=0 in this half-wave
#pragma unroll
    for (int nt = 0; nt < 8; ++nt) {
        int k  = n_wave + nt * 16 + M;
        int c  = k >> 6;
        int kh = (k >> 3) & 7;
        int kw = k & 7;
        size_t o0 = (((size_t)(b * CCH + c) * HH) + (size_t)(hp * 8 + kh)) * WW
                  + (size_t)wp0 * 8 + kw;
#pragma unroll
        for (int v = 0; v < 8; ++v) {
            out[o0 + (size_t)v * 8] = acc[nt][v];
        }
    }
}

// ---------------------------------------------------------------------------
extern "C" void kernel_launch(void* const* d_in, const int* in_sizes, int n_in,
                              void* d_out, int out_size, void* d_ws, size_t ws_size,
                              hipStream_t stream)
{
    const float* x    = (const float*)d_in[0];   // [8,16,512,512]
    const float* attn = (const float*)d_in[1];   // [8,1,4096,64]
    float* out = (float*)d_out;                  // [8,16,512,512]
    float* U2  = (float*)d_ws;                   // [8][32][1024][2] = 2 MB

    // 8*16*64*64 = 524288 pooled elements / 256 threads
    pool_pack_kernel<<<2048, 256, 0, stream>>>(x, U2);
    // 8 batches * 256 M-tiles
    att_gemm_kernel<<<2048, 256, 0, stream>>>(attn, U2, out);
}